// Factorization_44899588113111
// MI455X (gfx1250) — compile-verified
//
#include <hip/hip_runtime.h>
#include <cstdint>
#include <math.h>

#define HWSZ    262144            // 512*512
#define IMG     (3*HWSZ)          // per-batch elements
#define BATCH   16
#define NFACT   4
#define MAXITER 8
#define EPSF    1.1920928955078125e-07f
#define BPB     256               // blocks per batch (field kernels)
#define TPB     256               // threads per block; 4 pixels/thread

// ---------- helpers: 3-channel x 4-pixel tile, b128 loads/stores ----------
__device__ __forceinline__ void loadT(const float* __restrict__ base, int off, float v[3][4]) {
#pragma unroll
  for (int c = 0; c < 3; ++c) {
    float4 t = *reinterpret_cast<const float4*>(base + c*HWSZ + off);
    v[c][0]=t.x; v[c][1]=t.y; v[c][2]=t.z; v[c][3]=t.w;
  }
}
__device__ __forceinline__ void storeT(float* __restrict__ base, int off, const float v[3][4]) {
#pragma unroll
  for (int c = 0; c < 3; ++c) {
    *reinterpret_cast<float4*>(base + c*HWSZ + off) =
        make_float4(v[c][0], v[c][1], v[c][2], v[c][3]);
  }
}

__device__ __forceinline__ void blockReduceStore(float v, float* sm, float* dst) {
  const int t = threadIdx.x;
  sm[t] = v;
  __syncthreads();
#pragma unroll
  for (int s = TPB/2; s > 0; s >>= 1) {
    if (t < s) sm[t] += sm[t + s];
    __syncthreads();
  }
  if (t == 0) *dst = sm[0];
  __syncthreads();   // allow sm reuse
}

// ---------- factor init: E0 = thresE(X), partials for ||X-E0|| and ||X||^2 ----------
__global__ __launch_bounds__(TPB) void k_init(
    const float* __restrict__ X, float* __restrict__ E,
    float* __restrict__ partA, float* __restrict__ partX2,
    const float* __restrict__ lE, const float* __restrict__ st, int fi)
{
  __shared__ float sm[TPB];
  const int b   = blockIdx.y;
  const int off = (blockIdx.x * TPB + threadIdx.x) * 4;
  const float thE0 = lE[fi*MAXITER] / st[fi*MAXITER];
  float x[3][4], e[3][4];
  loadT(X + (size_t)b*IMG, off, x);
  float accA = 0.f, accS = 0.f;
#pragma unroll
  for (int p = 0; p < 4; ++p) {
    float n2 = x[0][p]*x[0][p] + x[1][p]*x[1][p] + x[2][p]*x[2][p];
    accS += n2;
    float n  = sqrtf(n2);
    float sE = fmaxf(1.f - thE0 / (n + EPSF), 0.f);
    float pn2 = 0.f;
#pragma unroll
    for (int c = 0; c < 3; ++c) {
      float ev = sE * x[c][p];
      e[c][p]  = ev;
      float pv = x[c][p] - ev;
      pn2 += pv*pv;
    }
    accA += sqrtf(pn2);
  }
  storeT(E + (size_t)b*IMG, off, e);
  blockReduceStore(accA, sm, &partA[b*BPB + blockIdx.x]);
  blockReduceStore(accS, sm, &partX2[b*BPB + blockIdx.x]);
}

// ---------- reductions: blocks 0..15 -> per-batch sA ; block 16 -> X2 denom ----------
__global__ __launch_bounds__(TPB) void k_reduce(
    const float* __restrict__ partA, const float* __restrict__ partX2,
    float* __restrict__ red,
    const float* __restrict__ lA, const float* __restrict__ st, int fi, int t)
{
  __shared__ float sm[TPB];
  const int tid = threadIdx.x;
  if (blockIdx.x < BATCH) {
    const int b = blockIdx.x;
    const float* src = partA + b * BPB;
    // CDNA5 async copy: 256 partials -> LDS (ASYNCcnt-tracked), then tree-reduce.
    unsigned ldsOff = (unsigned)(uintptr_t)(&sm[tid]);  // low 32b of flat LDS ptr = LDS byte offset
    unsigned vOff   = tid * 4u;
    asm volatile("global_load_async_to_lds_b32 %0, %1, %2"
                 :: "v"(ldsOff), "v"(vOff), "s"(src)
                 : "memory");
    asm volatile("s_wait_asynccnt 0x0" ::: "memory");
    __syncthreads();
#pragma unroll
    for (int s = TPB/2; s > 0; s >>= 1) {
      if (tid < s) sm[tid] += sm[tid + s];
      __syncthreads();
    }
    if (tid == 0) {
      float nn  = sqrtf(sm[0] + EPSF);
      float thA = lA[fi*MAXITER + t] / st[fi*MAXITER + t];
      red[b] = fmaxf(1.f - thA / (nn + EPSF), 0.f);
    }
  } else {
    // global ||X||: sum all BATCH*BPB squared-partials
    float v = 0.f;
    for (int j = tid; j < BATCH*BPB; j += TPB) v += partX2[j];
    sm[tid] = v;
    __syncthreads();
#pragma unroll
    for (int s = TPB/2; s > 0; s >>= 1) {
      if (tid < s) sm[tid] += sm[tid + s];
      __syncthreads();
    }
    if (tid == 0) red[BATCH] = sqrtf(sm[0]) + EPSF;   // denominator (X2 + EPS)
  }
}

// ---------- fused: A0,Y0 (registers) -> E1 + partials ----------
__global__ __launch_bounds__(TPB) void k_g0(
    const float* __restrict__ X, float* __restrict__ E, float* __restrict__ Y,
    float* __restrict__ partA, const float* __restrict__ red,
    const float* __restrict__ lE, const float* __restrict__ st, int fi)
{
  __shared__ float sm[TPB];
  const int b   = blockIdx.y;
  const int off = (blockIdx.x * TPB + threadIdx.x) * 4;
  const float sA   = red[b];
  const float den  = red[BATCH];            // X2 + EPS
  const float st1  = st[fi*MAXITER + 1];
  const float thE1 = lE[fi*MAXITER + 1] / st1;
  float x[3][4], e[3][4], y[3][4];
  loadT(X + (size_t)b*IMG, off, x);
  loadT(E + (size_t)b*IMG, off, e);
  float acc = 0.f;
#pragma unroll
  for (int p = 0; p < 4; ++p) {
    float q[3], yd[3];
#pragma unroll
    for (int c = 0; c < 3; ++c) {
      float a  = sA * (x[c][p] - e[c][p]);  // A0 in registers, never stored
      float y0 = x[c][p] / den;             // Y0 = X/(X2+EPS)
      y[c][p]  = y0;
      yd[c]    = y0 / st1;
      q[c]     = x[c][p] - a - yd[c];
    }
    float nq = sqrtf(q[0]*q[0] + q[1]*q[1] + q[2]*q[2]);
    float sE = fmaxf(1.f - thE1 / (nq + EPSF), 0.f);
    float pn2 = 0.f;
#pragma unroll
    for (int c = 0; c < 3; ++c) {
      float en = sE * q[c];                 // E1
      e[c][p]  = en;
      float pv = x[c][p] - en - yd[c];      // input to thresA at t=1
      pn2 += pv*pv;
    }
    acc += sqrtf(pn2);
  }
  storeT(E + (size_t)b*IMG, off, e);
  storeT(Y + (size_t)b*IMG, off, y);
  blockReduceStore(acc, sm, &partA[b*BPB + blockIdx.x]);
}

// ---------- fused iteration: A_t (regs), Y_t, E_{t+1} + partials ----------
__global__ __launch_bounds__(TPB) void k_giter(
    const float* __restrict__ X, float* __restrict__ E, float* __restrict__ Y,
    float* __restrict__ partA, const float* __restrict__ red,
    const float* __restrict__ lE, const float* __restrict__ st,
    int fi, int t, int last)
{
  __shared__ float sm[TPB];
  const int b   = blockIdx.y;
  const int off = (blockIdx.x * TPB + threadIdx.x) * 4;
  const float sA   = red[b];
  const float stt  = st[fi*MAXITER + t];
  const float stn  = st[fi*MAXITER + t + 1];
  const float thEn = lE[fi*MAXITER + t + 1] / stn;
  float x[3][4], e[3][4], y[3][4];
  loadT(X + (size_t)b*IMG, off, x);
  loadT(E + (size_t)b*IMG, off, e);
  loadT(Y + (size_t)b*IMG, off, y);
  float acc = 0.f;
#pragma unroll
  for (int p = 0; p < 4; ++p) {
    float q[3], yn[3], yd[3];
#pragma unroll
    for (int c = 0; c < 3; ++c) {
      float pv = x[c][p] - e[c][p] - y[c][p] / stt;  // thresA input at t
      float a  = sA * pv;                            // A_t (registers only)
      float yv = y[c][p] + stt * (e[c][p] + a - x[c][p]);  // Y_t
      yn[c] = yv;
      yd[c] = yv / stn;
      q[c]  = x[c][p] - a - yd[c];                   // thresE input at t+1
    }
    float nq = sqrtf(q[0]*q[0] + q[1]*q[1] + q[2]*q[2]);
    float sE = fmaxf(1.f - thEn / (nq + EPSF), 0.f);
    float pn2 = 0.f;
#pragma unroll
    for (int c = 0; c < 3; ++c) {
      float en = sE * q[c];                          // E_{t+1}
      e[c][p]  = en;
      y[c][p]  = yn[c];
      float pv = x[c][p] - en - yd[c];
      pn2 += pv*pv;
    }
    acc += sqrtf(pn2);
  }
  storeT(E + (size_t)b*IMG, off, e);
  if (!last) {                    // t==6 produces E7; A7/Y7 are dead in the reference
    storeT(Y + (size_t)b*IMG, off, y);
    blockReduceStore(acc, sm, &partA[b*BPB + blockIdx.x]);
  }
}

// ---------- per-factor epilogue: e=relu(E7); out_i = e or |e - out_{i-1}|; Xf -= e ----------
__global__ __launch_bounds__(TPB) void k_final(
    const float* __restrict__ E, const float* __restrict__ XfSrc,
    float* __restrict__ XfDst, float* __restrict__ out, int fi, int writeXf)
{
  const int b   = blockIdx.y;
  const int off = (blockIdx.x * TPB + threadIdx.x) * 4;
  float e[3][4];
  loadT(E + (size_t)b*IMG, off, e);
#pragma unroll
  for (int c = 0; c < 3; ++c)
#pragma unroll
    for (int p = 0; p < 4; ++p) e[c][p] = fmaxf(e[c][p], 0.f);

  if (writeXf) {
    float x[3][4];
    loadT(XfSrc + (size_t)b*IMG, off, x);
#pragma unroll
    for (int c = 0; c < 3; ++c)
#pragma unroll
      for (int p = 0; p < 4; ++p) x[c][p] -= e[c][p];
    storeT(XfDst + (size_t)b*IMG, off, x);
  }

  float* outB = out + (size_t)b * 12 * HWSZ;
  if (fi == 0) {
    storeT(outB, off, e);
  } else {
    float prev[3][4];
    loadT(outB + (size_t)(fi-1)*3*HWSZ, off, prev);
#pragma unroll
    for (int c = 0; c < 3; ++c)
#pragma unroll
      for (int p = 0; p < 4; ++p) e[c][p] = fabsf(e[c][p] - prev[c][p]);
    storeT(outB + (size_t)fi*3*HWSZ, off, e);
  }
}

__global__ void k_loss(float* __restrict__ out, size_t idx) { out[idx] = 0.f; }

// ---------- host launcher ----------
extern "C" void kernel_launch(void* const* d_in, const int* in_sizes, int n_in,
                              void* d_out, int out_size, void* d_ws, size_t ws_size,
                              hipStream_t stream)
{
  (void)in_sizes; (void)n_in; (void)out_size; (void)ws_size;
  const float* x  = (const float*)d_in[0];
  const float* lA = (const float*)d_in[1];
  const float* lE = (const float*)d_in[2];
  const float* st = (const float*)d_in[3];
  float* out = (float*)d_out;

  // workspace: Xf, E, Y fields + partials + scalars  (~151 MB)
  float* Xf     = (float*)d_ws;
  float* E      = Xf + (size_t)BATCH * IMG;
  float* Y      = E  + (size_t)BATCH * IMG;
  float* partA  = Y  + (size_t)BATCH * IMG;
  float* partX2 = partA  + BATCH * BPB;
  float* red    = partX2 + BATCH * BPB;   // [0..15]=sA per batch, [16]=X2+EPS

  dim3 grid(BPB, BATCH);
  k_loss<<<1, 1, 0, stream>>>(out, (size_t)BATCH * 12 * HWSZ);

  for (int fi = 0; fi < NFACT; ++fi) {
    const float* Xs = (fi == 0) ? x : Xf;
    k_init  <<<grid, TPB, 0, stream>>>(Xs, E, partA, partX2, lE, st, fi);
    k_reduce<<<BATCH + 1, TPB, 0, stream>>>(partA, partX2, red, lA, st, fi, 0);
    k_g0    <<<grid, TPB, 0, stream>>>(Xs, E, Y, partA, red, lE, st, fi);
    for (int t = 1; t <= 6; ++t) {
      k_reduce<<<BATCH, TPB, 0, stream>>>(partA, partX2, red, lA, st, fi, t);
      k_giter <<<grid, TPB, 0, stream>>>(Xs, E, Y, partA, red, lE, st, fi, t, (t == 6));
    }
    k_final<<<grid, TPB, 0, stream>>>(E, Xs, Xf, out, fi, (fi < 3));
  }
}